// diff_simple_TF_75788992905245
// MI455X (gfx1250) — compile-verified
//
#include <hip/hip_runtime.h>
#include <hip/hip_bf16.h>

typedef __attribute__((ext_vector_type(2))) float v2f;
typedef __attribute__((ext_vector_type(8))) float v8f;

// ---------------------------------------------------------------------------
// Kernel 1: zero the whole output region (rel[B] ++ d[Vp1*B]).
// ~414 MB of stores -> this is the HBM-bandwidth-dominant phase.
// ---------------------------------------------------------------------------
__global__ void zero_out_kernel(float* __restrict__ p, size_t n) {
  size_t i = (size_t)blockIdx.x * blockDim.x + threadIdx.x;
  size_t stride = (size_t)gridDim.x * blockDim.x;
  size_t n4 = n >> 2;
  float4 z = make_float4(0.f, 0.f, 0.f, 0.f);
  float4* __restrict__ p4 = (float4*)p;
  for (size_t j = i; j < n4; j += stride) p4[j] = z;
  size_t tail = n4 << 2;
  size_t rem = n - tail;
  if (i < rem) p[tail + i] = 0.f;
}

// ---------------------------------------------------------------------------
// Kernel 2: per-vocab TDV weight via V_WMMA_F32_16X16X4_F32.
//   tdv[v] = relu(emb[v,:] . W + b)
// One wave handles 16 vocab rows. A tile = 16x4 chunk of emb
// (ISA layout: lanes 0-15 carry K=0(v0),K=1(v1); lanes 16-31 carry K=2,K=3).
// B tile = W[k..k+3] broadcast across all 16 columns (same K-split layout),
// so every column of the 16x16 f32 accumulator holds the same dot product.
// 75 WMMA steps cover K=300. Lanes 0 and 16 write rows 0-7 / 8-15.
// ---------------------------------------------------------------------------
__global__ void tdv_wmma_kernel(const float* __restrict__ emb,
                                const float* __restrict__ W,
                                const float* __restrict__ bias,
                                float* __restrict__ tdv,
                                int Vp1, int Dd) {
  const int lane = threadIdx.x & 31;
  const int tile = blockIdx.x * (blockDim.x >> 5) + (threadIdx.x >> 5);
  const int base = tile * 16;
  if (base >= Vp1) return;                 // wave-uniform exit, EXEC stays full

  int row = base + (lane & 15);
  row = row < Vp1 ? row : Vp1 - 1;         // per-lane clamp (select, not branch)
  const int khalf = (lane >> 4) << 1;      // 0 for lanes 0-15, 2 for lanes 16-31
  const float* __restrict__ arow = emb + (size_t)row * Dd;

  v8f c = {};
  for (int k0 = 0; k0 + 3 < Dd; k0 += 4) {
    v2f a, bm;
    a.x  = arow[k0 + khalf];
    a.y  = arow[k0 + khalf + 1];
    bm.x = W[k0 + khalf];
    bm.y = W[k0 + khalf + 1];
    // (neg_a, A, neg_b, B, c_mod, C, reuse_a, reuse_b)
    c = __builtin_amdgcn_wmma_f32_16x16x4_f32(false, a, false, bm, (short)0, c,
                                              false, false);
  }

  const float bb = bias[0];
  if (lane == 0 || lane == 16) {           // columns identical -> one lane per half
    const int r0 = base + ((lane >> 4) << 3);
#pragma unroll
    for (int i = 0; i < 8; ++i) {
      const int r = r0 + i;
      if (r < Vp1) {
        float x = c[i] + bb;
        tdv[r] = x > 0.f ? x : 0.f;        // relu
      }
    }
  }
}

// ---------------------------------------------------------------------------
// Kernel 3: scatter doc values into dense d:  d[tok, col] += tdv[tok]*freq
// ---------------------------------------------------------------------------
__global__ void doc_scatter_kernel(const int2* __restrict__ d_idx,
                                   const float* __restrict__ d_freq,
                                   const float* __restrict__ tdv,
                                   float* __restrict__ dmat,
                                   int ND, int B) {
  int i = blockIdx.x * blockDim.x + threadIdx.x;
  if (i >= ND) return;
  int2 tc = d_idx[i];                      // {token, doc_col}
  float v = tdv[tc.x] * d_freq[i];
  atomicAdd(dmat + (size_t)tc.x * B + tc.y, v);
}

// ---------------------------------------------------------------------------
// Kernel 4: rel[col] = sum over sparse q of q_freq * d[tok, col]
// (q*d is linear in q -> never materialize dense q)
// ---------------------------------------------------------------------------
__global__ void q_reduce_kernel(const int2* __restrict__ q_idx,
                                const float* __restrict__ q_freq,
                                const float* __restrict__ dmat,
                                float* __restrict__ rel,
                                int NQ, int B) {
  int i = blockIdx.x * blockDim.x + threadIdx.x;
  if (i >= NQ) return;
  int2 tc = q_idx[i];                      // {token, query_col}
  atomicAdd(rel + tc.y, q_freq[i] * dmat[(size_t)tc.x * B + tc.y]);
}

// ---------------------------------------------------------------------------
extern "C" void kernel_launch(void* const* d_in, const int* in_sizes, int n_in,
                              void* d_out, int out_size, void* d_ws, size_t ws_size,
                              hipStream_t stream) {
  const int2*  q_idx  = (const int2*)d_in[0];   // [NQ,2]
  const float* q_freq = (const float*)d_in[1];  // [NQ]
  const int2*  d_idx  = (const int2*)d_in[2];   // [ND,2]
  // d_in[3] = d_indices_bow: redundant (== d_idx[:,0]), unused
  const float* d_freq = (const float*)d_in[4];  // [ND]
  const float* emb    = (const float*)d_in[5];  // [Vp1, Dd]
  const float* W      = (const float*)d_in[6];  // [Dd]
  const float* bias   = (const float*)d_in[7];  // [1]
  // d_in[8] = batch_size scalar (device side); derive B on host instead.

  const int NQ  = in_sizes[1];
  const int ND  = in_sizes[4];
  const int Dd  = in_sizes[6];              // 300
  const int Vp1 = in_sizes[5] / Dd;         // 100001
  const int B   = out_size / (Vp1 + 1);     // rel[B] + d[Vp1*B]

  float* rel  = (float*)d_out;              // first B floats
  float* dmat = (float*)d_out + B;          // then Vp1*B floats
  float* tdv  = (float*)d_ws;               // Vp1 floats of scratch

  // 1) zero rel + dense d (must re-zero every call; atomics accumulate)
  zero_out_kernel<<<8192, 256, 0, stream>>>((float*)d_out, (size_t)out_size);

  // 2) per-vocab TDV term weights via WMMA (one wave per 16 rows)
  const int tiles  = (Vp1 + 15) / 16;
  const int blocks = (tiles + 7) / 8;       // 8 waves (256 threads) per block
  tdv_wmma_kernel<<<blocks, 256, 0, stream>>>(emb, W, bias, tdv, Vp1, Dd);

  // 3) scatter docs into dense d
  doc_scatter_kernel<<<(ND + 255) / 256, 256, 0, stream>>>(d_idx, d_freq, tdv,
                                                           dmat, ND, B);

  // 4) sparse q . dense d -> rel
  q_reduce_kernel<<<(NQ + 255) / 256, 256, 0, stream>>>(q_idx, q_freq, dmat,
                                                        rel, NQ, B);
}